// LiftedStructureLoss_57629871178316
// MI455X (gfx1250) — compile-verified
//
#include <hip/hip_runtime.h>
#include <hip/hip_bf16.h>

// Lifted-structure loss, B=4096, D=128, fused WMMA-f32 pipeline (MI455X / gfx1250).
//
// Pass A: mag[i] = ||x_i||^2                      (one wave per row, b128 loads)
// Pass 1: row_neg[i] via 16x16 sim tiles:         V_WMMA_F32_16X16X4_F32, B tile
//         staged in LDS per workgroup (8 waves share it), double-buffered with
//         GLOBAL_LOAD_ASYNC_TO_LDS_B128 prefetch (ASYNCcnt-tracked).
// Pass 2: recompute sim tiles for j-tiles >= block's first strip, fused
//         log/hinge^2 + positive-pair count, per-wave partials to scratch.
// Pass 3: deterministic serial reduction of 256 partials -> scalar loss.

typedef __attribute__((ext_vector_type(2))) float v2f;
typedef __attribute__((ext_vector_type(8))) float v8f;

#define BSZ    4096
#define DIM    128
#define NTILE  (BSZ / 16)   // 256 column tiles of 16
#define MARGIN 1.0f
#define ROWP   132          // LDS row pitch (floats): 132 % 64 == 4 -> conflict-free frag reads

// gfx1250 async global->LDS path (ASYNCcnt-tracked), with portable fallback.
#if defined(__gfx1250__) && __has_builtin(__builtin_amdgcn_global_load_async_to_lds_b128)
#define HAVE_ASYNC_LDS 1
#else
#define HAVE_ASYNC_LDS 0
#endif

#if HAVE_ASYNC_LDS
// The builtin takes pointers to 16-byte int vectors in the global (AS1) and
// LDS (AS3) address spaces (per hipcc's diagnostic of the parameter type).
typedef int v4i_b128 __attribute__((__vector_size__(4 * sizeof(int))));
typedef __attribute__((address_space(1))) v4i_b128* gas_v4i;
typedef __attribute__((address_space(3))) v4i_b128* las_v4i;
#endif

__device__ __forceinline__ void wait_async_all() {
#if HAVE_ASYNC_LDS
#if __has_builtin(__builtin_amdgcn_s_wait_asynccnt)
    __builtin_amdgcn_s_wait_asynccnt(0);
#else
    asm volatile("s_wait_asynccnt 0x0" ::: "memory");
#endif
#endif
}

// Cooperative stage of one 16x128 B tile into LDS (256 threads, 2 x b128 each).
// Coalesced: consecutive lanes fetch consecutive 16B chunks of one row.
__device__ __forceinline__ void stage_tile(float* dstLds, const float* __restrict__ X,
                                           int jBase) {
    const int t = (int)threadIdx.x;
#pragma unroll
    for (int k = 0; k < 2; ++k) {
        const int c    = t + k * 256;     // 512 chunks total
        const int row  = c >> 5;          // 0..15
        const int col  = (c & 31) * 4;    // float4 column
#if HAVE_ASYNC_LDS
        __builtin_amdgcn_global_load_async_to_lds_b128(
            (gas_v4i)(X + (size_t)(jBase + row) * DIM + col),
            (las_v4i)(dstLds + row * ROWP + col),
            0, 0);
#else
        *(float4*)(dstLds + row * ROWP + col) =
            *(const float4*)(X + (size_t)(jBase + row) * DIM + col);
#endif
    }
}

// ---------------- Pass A: squared magnitudes (one wave per row) ----------------
__global__ void lsl_mag_kernel(const float* __restrict__ X, float* __restrict__ mag) {
    const int wave = (int)((blockIdx.x * blockDim.x + threadIdx.x) >> 5);
    const int lane = (int)(threadIdx.x & 31);
    if (wave >= BSZ) return;
    const float4 v = *(const float4*)(X + (size_t)wave * DIM + lane * 4);
    float s = v.x * v.x + v.y * v.y + v.z * v.z + v.w * v.w;
#pragma unroll
    for (int m = 1; m < 32; m <<= 1) s += __shfl_xor(s, m, 32);
    if (lane == 0) mag[wave] = s;
}

// ---------------- Pass 1: row_neg via WMMA sim tiles ----------------
// 256 threads = 8 waves; wave w owns rows [ (blk*8+w)*16 , +16 ).
__global__ void lsl_rowneg_kernel(const float* __restrict__ X,
                                  const long long* __restrict__ T,
                                  const float* __restrict__ mag,
                                  float* __restrict__ rowneg) {
    __shared__ float sB[2][16 * ROWP];

    const int lane  = (int)(threadIdx.x & 31);
    const int wave  = (int)(blockIdx.x * 8 + (threadIdx.x >> 5)); // 0..255
    const int rBase = wave * 16;
    const int h     = lane >> 4;   // lane half
    const int l15   = lane & 15;

    // A fragments for all 32 k-steps: lane holds row rBase+l15, K = 4s+2h..+1.
    v2f afrag[32];
    const float* arow = X + (size_t)(rBase + l15) * DIM + 2 * h;
#pragma unroll
    for (int s = 0; s < 32; ++s) afrag[s] = *(const v2f*)(arow + 4 * s);

    float     magM[8];
    long long tM[8];
#pragma unroll
    for (int r = 0; r < 8; ++r) {
        magM[r] = mag[rBase + r + 8 * h];
        tM[r]   = T[rBase + r + 8 * h];
    }

    float acc[8] = {0.f, 0.f, 0.f, 0.f, 0.f, 0.f, 0.f, 0.f};

    stage_tile(sB[0], X, 0);
    wait_async_all();
    __syncthreads();

    int p = 0;
    for (int jt = 0; jt < NTILE; ++jt) {
        if (jt + 1 < NTILE) stage_tile(sB[p ^ 1], X, (jt + 1) * 16);  // prefetch overlap

        const float* bbase = &sB[p][l15 * ROWP + 2 * h];
        v8f c = {};
#pragma unroll
        for (int g = 0; g < 4; ++g) {            // batches of 8: amortize ds waits
            v2f bf[8];
#pragma unroll
            for (int u = 0; u < 8; ++u) bf[u] = *(const v2f*)(bbase + 4 * (8 * g + u));
#pragma unroll
            for (int u = 0; u < 8; ++u)
                c = __builtin_amdgcn_wmma_f32_16x16x4_f32(
                        false, afrag[8 * g + u], false, bf[u], (short)0, c, false, false);
        }

        const int       jBase = jt * 16;
        const float     magN  = mag[jBase + l15];
        const long long tN    = T[jBase + l15];
#pragma unroll
        for (int r = 0; r < 8; ++r) {
            const float d2   = magM[r] + magN - 2.0f * c[r];
            const float dist = sqrtf(fmaxf(d2, 0.0f));
            if (tM[r] != tN) acc[r] += __expf(MARGIN - dist);
        }

        wait_async_all();
        __syncthreads();
        p ^= 1;
    }

    // Reduce across the 16 lanes of each half (rows r and r+8 independently).
#pragma unroll
    for (int r = 0; r < 8; ++r) {
        float s = acc[r];
        s += __shfl_xor(s, 1, 32);
        s += __shfl_xor(s, 2, 32);
        s += __shfl_xor(s, 4, 32);
        s += __shfl_xor(s, 8, 32);
        if (l15 == 0) rowneg[rBase + r + 8 * h] = s;
    }
}

// ---------------- Pass 2: hinge^2 over positive pairs (i<j) ----------------
__global__ void lsl_loss_kernel(const float* __restrict__ X,
                                const long long* __restrict__ T,
                                const float* __restrict__ mag,
                                const float* __restrict__ rowneg,
                                float* __restrict__ psum,
                                unsigned* __restrict__ pcnt) {
    __shared__ float sB[2][16 * ROWP];

    const int lane  = (int)(threadIdx.x & 31);
    const int wave  = (int)(blockIdx.x * 8 + (threadIdx.x >> 5)); // 0..255
    const int rBase = wave * 16;
    const int h     = lane >> 4;
    const int l15   = lane & 15;

    v2f afrag[32];
    const float* arow = X + (size_t)(rBase + l15) * DIM + 2 * h;
#pragma unroll
    for (int s = 0; s < 32; ++s) afrag[s] = *(const v2f*)(arow + 4 * s);

    float     magM[8], rnM[8];
    long long tM[8];
    int       M[8];
#pragma unroll
    for (int r = 0; r < 8; ++r) {
        M[r]    = rBase + r + 8 * h;
        magM[r] = mag[M[r]];
        rnM[r]  = rowneg[M[r]];
        tM[r]   = T[M[r]];
    }

    float    sum = 0.0f;
    unsigned cnt = 0u;

    // All 8 waves share the j loop; start at the block's first strip so every
    // (i<j) pair is covered. The M<N mask discards the lower triangle.
    const int jt0 = (int)blockIdx.x * 8;

    stage_tile(sB[0], X, jt0 * 16);
    wait_async_all();
    __syncthreads();

    int p = 0;
    for (int jt = jt0; jt < NTILE; ++jt) {
        if (jt + 1 < NTILE) stage_tile(sB[p ^ 1], X, (jt + 1) * 16);

        const float* bbase = &sB[p][l15 * ROWP + 2 * h];
        v8f c = {};
#pragma unroll
        for (int g = 0; g < 4; ++g) {
            v2f bf[8];
#pragma unroll
            for (int u = 0; u < 8; ++u) bf[u] = *(const v2f*)(bbase + 4 * (8 * g + u));
#pragma unroll
            for (int u = 0; u < 8; ++u)
                c = __builtin_amdgcn_wmma_f32_16x16x4_f32(
                        false, afrag[8 * g + u], false, bf[u], (short)0, c, false, false);
        }

        const int       N    = jt * 16 + l15;
        const float     magN = mag[N];
        const float     rnN  = rowneg[N];
        const long long tN   = T[N];
#pragma unroll
        for (int r = 0; r < 8; ++r) {
            if ((M[r] < N) && (tM[r] == tN)) {
                const float d2   = magM[r] + magN - 2.0f * c[r];
                const float dist = sqrtf(fmaxf(d2, 0.0f));
                const float ln   = __logf(rnM[r] + rnN);
                const float hng  = fmaxf(ln + dist, 0.0f);
                sum += hng * hng;
                cnt += 1u;
            }
        }

        wait_async_all();
        __syncthreads();
        p ^= 1;
    }

    // Full-wave reduction (fixed order -> deterministic).
#pragma unroll
    for (int m = 1; m < 32; m <<= 1) {
        sum += __shfl_xor(sum, m, 32);
        cnt += __shfl_xor(cnt, m, 32);
    }
    if (lane == 0) {
        psum[wave] = sum;
        pcnt[wave] = cnt;
    }
}

// ---------------- Pass 3: deterministic final reduction ----------------
__global__ void lsl_final_kernel(const float* __restrict__ psum,
                                 const unsigned* __restrict__ pcnt,
                                 float* __restrict__ out) {
    if (threadIdx.x == 0 && blockIdx.x == 0) {
        float    s = 0.0f;
        unsigned c = 0u;
        for (int i = 0; i < NTILE; ++i) { s += psum[i]; c += pcnt[i]; }
        out[0] = s / (2.0f * (float)c);
    }
}

extern "C" void kernel_launch(void* const* d_in, const int* in_sizes, int n_in,
                              void* d_out, int out_size, void* d_ws, size_t ws_size,
                              hipStream_t stream) {
    (void)in_sizes; (void)n_in; (void)out_size; (void)ws_size;
    const float*     X   = (const float*)d_in[0];
    const long long* T   = (const long long*)d_in[1];
    float*           out = (float*)d_out;

    // Workspace layout (all fully overwritten every call):
    float*    mag    = (float*)d_ws;              // 4096 floats
    float*    rowneg = mag + BSZ;                 // 4096 floats
    float*    psum   = rowneg + BSZ;              // 256 floats
    unsigned* pcnt   = (unsigned*)(psum + NTILE); // 256 uints

    // Pass A: 4096 waves (one per row), 8 waves per block.
    lsl_mag_kernel<<<(BSZ * 32) / 256, 256, 0, stream>>>(X, mag);

    // Pass 1 / Pass 2: 256 waves = 8 strips per block, 32 blocks.
    lsl_rowneg_kernel<<<NTILE / 8, 256, 0, stream>>>(X, T, mag, rowneg);
    lsl_loss_kernel<<<NTILE / 8, 256, 0, stream>>>(X, T, mag, rowneg, psum, pcnt);

    // Pass 3: scalar finish.
    lsl_final_kernel<<<1, 32, 0, stream>>>(psum, pcnt, out);
}